// FastTSAGEConv_8821862826162
// MI455X (gfx1250) — compile-verified
//
#include <hip/hip_runtime.h>
#include <hip/hip_bf16.h>

// ---------------------------------------------------------------------------
// FastTSAGEConv for MI455X (gfx1250, wave32, WMMA)
//   out = dst_feat @ W_self + b_self + segcummean(src_feat, dst) @ W_neigh + b_neigh
// E=500000, D=DOUT=128.
// ---------------------------------------------------------------------------

#define E_EDGES 500000
#define DK      128
#define DOUTK   128
#define ROWS    64                       // rows per chunk (fused kernel tile)
#define NCHUNK  ((E_EDGES + ROWS - 1) / ROWS)   // 7813
#define SUP     64                       // chunks per superchunk
#define NSUPER  ((NCHUNK + SUP - 1) / SUP)      // 123

typedef __attribute__((ext_vector_type(2))) float v2f;
typedef __attribute__((ext_vector_type(8))) float v8f;

// ---------------------------------------------------------------------------
// Kernel 1: per-chunk segmented-scan summary.
//  trail[c][d] = sum of src rows from the last in-chunk segment start to chunk end
//  last_start[c] = global row index of last segment start inside chunk, or -1
// ---------------------------------------------------------------------------
__global__ __launch_bounds__(128)
void k_chunk_stats(const float* __restrict__ src, const int* __restrict__ dsti,
                   float* __restrict__ trail, int* __restrict__ last_start) {
    const int c  = blockIdx.x;
    const int r0 = c * ROWS;
    const int rend = min(r0 + ROWS, E_EDGES);
    const int d = threadIdx.x;            // feature column 0..127

    __shared__ int s_last;
    if (threadIdx.x == 0) s_last = -1;
    __syncthreads();

    // find last boundary in [r0, rend)
    for (int r = r0 + threadIdx.x; r < rend; r += 128) {
        const bool bnd = (r == 0) || (dsti[r] != dsti[r - 1]);
        if (bnd) atomicMax(&s_last, r);
    }
    __syncthreads();

    const int last = s_last;
    const int from = (last < 0) ? r0 : last;
    float acc = 0.0f;
    for (int r = from; r < rend; ++r)
        acc += src[(size_t)r * DK + d];

    trail[(size_t)c * DK + d] = acc;
    if (d == 0) last_start[c] = last;
}

// ---------------------------------------------------------------------------
// Kernel 2: combine SUP chunk summaries into one superchunk summary.
// Monoid: (sum, lastStart);  combine(x, y) = y.ls>=0 ? y : (x.sum+y.sum, x.ls)
// ---------------------------------------------------------------------------
__global__ __launch_bounds__(128)
void k_scan_super(const float* __restrict__ trail, const int* __restrict__ last_start,
                  float* __restrict__ strail, int* __restrict__ slast) {
    const int s = blockIdx.x;
    const int d = threadIdx.x;
    const int c0 = s * SUP;
    const int c1 = min(c0 + SUP, NCHUNK);
    float sum = 0.0f;
    int   ls  = -1;
    for (int c = c0; c < c1; ++c) {
        const float t = trail[(size_t)c * DK + d];
        const int   l = last_start[c];
        if (l >= 0) { sum = t; ls = l; } else { sum += t; }
    }
    strail[(size_t)s * DK + d] = sum;
    if (d == 0) slast[s] = ls;
}

// ---------------------------------------------------------------------------
// Kernel 3: sequential scan over the NSUPER superchunks (123 iterations).
// Emits the carry (sum + segment-start row) entering each superchunk.
// ---------------------------------------------------------------------------
__global__ __launch_bounds__(128)
void k_scan_top(const float* __restrict__ strail, const int* __restrict__ slast,
                float* __restrict__ scarry, int* __restrict__ scstart) {
    const int d = threadIdx.x;
    float cs = 0.0f;
    int   st = 0;
    for (int s = 0; s < NSUPER; ++s) {
        scarry[(size_t)s * DK + d] = cs;
        if (d == 0) scstart[s] = st;
        const float t = strail[(size_t)s * DK + d];
        const int   l = slast[s];
        if (l >= 0) { cs = t; st = l; } else { cs += t; }
    }
}

// ---------------------------------------------------------------------------
// Kernel 4: within each superchunk, produce the per-chunk carries.
// ---------------------------------------------------------------------------
__global__ __launch_bounds__(128)
void k_scan_apply(const float* __restrict__ trail, const int* __restrict__ last_start,
                  const float* __restrict__ scarry, const int* __restrict__ scstart,
                  float* __restrict__ carry_sum, int* __restrict__ carry_start) {
    const int s = blockIdx.x;
    const int d = threadIdx.x;
    const int c0 = s * SUP;
    const int c1 = min(c0 + SUP, NCHUNK);
    float cs = scarry[(size_t)s * DK + d];
    int   st = scstart[s];
    for (int c = c0; c < c1; ++c) {
        carry_sum[(size_t)c * DK + d] = cs;
        if (d == 0) carry_start[c] = st;
        const float t = trail[(size_t)c * DK + d];
        const int   l = last_start[c];
        if (l >= 0) { cs = t; st = l; } else { cs += t; }
    }
}

// ---------------------------------------------------------------------------
// Kernel 5: fused segmented-cummean + double GEMM with V_WMMA_F32_16X16X4_F32.
// 128 threads = 4 waves; each wave owns a 16-row stripe of the 64-row chunk.
// LDS: h[64][130] f32 (~33 KB) -> 2 workgroups per WGP.
// ---------------------------------------------------------------------------
__global__ __launch_bounds__(128, 2)
void k_fused(const float* __restrict__ src, const float* __restrict__ dstf,
             const int* __restrict__ dsti,
             const float* __restrict__ Wself, const float* __restrict__ bself,
             const float* __restrict__ Wneigh, const float* __restrict__ bneigh,
             const float* __restrict__ carry_sum, const int* __restrict__ carry_start,
             float* __restrict__ out) {
    __shared__ float h[ROWS][DK + 2];     // +2 pad: LDS bank spread for A-frag reads
    __shared__ int   sdst[ROWS + 1];      // dst[r0-1 .. r0+nrows-1]

    const int c     = blockIdx.x;
    const int r0    = c * ROWS;
    const int nrows = min(ROWS, E_EDGES - r0);   // 64, or 32 for the last chunk
    const int tid   = threadIdx.x;

    for (int i = tid; i <= nrows; i += 128) {
        const int r = r0 - 1 + i;
        sdst[i] = (r >= 0) ? dsti[r] : -1;       // sentinel forces boundary at r==0
    }
    __syncthreads();

    // ---- Phase A: segmented cumulative mean into LDS (thread = feature col) --
    {
        const int d = tid;
        float run = carry_sum[(size_t)c * DK + d];
        int start = carry_start[c];
        for (int i = 0; i < nrows; ++i) {
            const int r = r0 + i;
            if (sdst[i + 1] != sdst[i]) { run = 0.0f; start = r; } // wave-uniform
            run += src[(size_t)r * DK + d];
            h[i][d] = run / (float)(r - start + 1);
        }
    }
    __syncthreads();

    // ---- Phase B: out[16 rows x 128] = A_self@Wself + A_neigh@Wneigh + bias --
    const int wave = tid >> 5;
    const int lane = tid & 31;
    const int half = lane >> 4;       // 0: lanes 0-15, 1: lanes 16-31
    const int ln   = lane & 15;
    const int m0   = wave * 16;
    if (m0 >= nrows) return;          // wave-uniform: EXEC stays all-ones for WMMA

    // A fragments per ISA 16x4 f32 layout: lane(ln) row M=ln, K = kt*4 + 2*half {+0,+1}
    v2f a_s[32], a_n[32];
    const int    arow = m0 + ln;
    const size_t grow = (size_t)(r0 + arow) * DK;
#pragma unroll
    for (int kt = 0; kt < 32; ++kt) {
        const int kk = kt * 4 + half * 2;
        a_n[kt] = *(const v2f*)&h[arow][kk];        // ds_load_b64
        a_s[kt] = *(const v2f*)&dstf[grow + kk];    // global_load_b64
    }

    for (int nt = 0; nt < 8; ++nt) {
        const int n = nt * 16 + ln;
        const float bias = bself[n] + bneigh[n];
        v8f acc = { bias, bias, bias, bias, bias, bias, bias, bias };
#pragma unroll
        for (int kt = 0; kt < 32; ++kt) {
            const int kk = kt * 4 + half * 2;
            // B fragment per 4x16 layout: lane holds B[kk][n], B[kk+1][n]
            v2f bs = { Wself[(size_t)kk * DOUTK + n],  Wself[(size_t)(kk + 1) * DOUTK + n] };
            acc = __builtin_amdgcn_wmma_f32_16x16x4_f32(
                      false, a_s[kt], false, bs, (short)0, acc, false, false);
            v2f bn = { Wneigh[(size_t)kk * DOUTK + n], Wneigh[(size_t)(kk + 1) * DOUTK + n] };
            acc = __builtin_amdgcn_wmma_f32_16x16x4_f32(
                      false, a_n[kt], false, bn, (short)0, acc, false, false);
        }
        // D layout: VGPR j -> M = j + 8*half, N = n
        const size_t obase = (size_t)(r0 + m0 + half * 8) * DOUTK + n;
#pragma unroll
        for (int j = 0; j < 8; ++j)
            out[obase + (size_t)j * DOUTK] = acc[j];
    }
}

// ---------------------------------------------------------------------------
extern "C" void kernel_launch(void* const* d_in, const int* in_sizes, int n_in,
                              void* d_out, int out_size, void* d_ws, size_t ws_size,
                              hipStream_t stream) {
    const float* src    = (const float*)d_in[0];   // [E,128]
    const float* dstf   = (const float*)d_in[1];   // [E,128]
    const int*   dsti   = (const int*)  d_in[2];   // [E]
    const float* Wself  = (const float*)d_in[3];   // [128,128]
    const float* bself  = (const float*)d_in[4];   // [128]
    const float* Wneigh = (const float*)d_in[5];   // [128,128]
    const float* bneigh = (const float*)d_in[6];   // [128]
    float*       out    = (float*)d_out;           // [E,128]

    char* ws = (char*)d_ws;
    float* trail      = (float*)ws;  ws += sizeof(float) * (size_t)NCHUNK * DK;
    float* carry_sum  = (float*)ws;  ws += sizeof(float) * (size_t)NCHUNK * DK;
    float* strail     = (float*)ws;  ws += sizeof(float) * (size_t)NSUPER * DK;
    float* scarry     = (float*)ws;  ws += sizeof(float) * (size_t)NSUPER * DK;
    int*   last_start = (int*)ws;    ws += sizeof(int) * (size_t)NCHUNK;
    int*   carry_start= (int*)ws;    ws += sizeof(int) * (size_t)NCHUNK;
    int*   slast      = (int*)ws;    ws += sizeof(int) * (size_t)NSUPER;
    int*   scstart    = (int*)ws;    ws += sizeof(int) * (size_t)NSUPER;

    k_chunk_stats<<<NCHUNK, 128, 0, stream>>>(src, dsti, trail, last_start);
    k_scan_super <<<NSUPER, 128, 0, stream>>>(trail, last_start, strail, slast);
    k_scan_top   <<<1,      128, 0, stream>>>(strail, slast, scarry, scstart);
    k_scan_apply <<<NSUPER, 128, 0, stream>>>(trail, last_start, scarry, scstart,
                                              carry_sum, carry_start);
    k_fused      <<<NCHUNK, 128, 0, stream>>>(src, dstf, dsti, Wself, bself,
                                              Wneigh, bneigh, carry_sum, carry_start, out);
}